// MaskAutoEncoder_25752623907310
// MI455X (gfx1250) — compile-verified
//
#include <hip/hip_runtime.h>
#include <hip/hip_bf16.h>
#include <math.h>

typedef __attribute__((ext_vector_type(2))) float v2f;
typedef __attribute__((ext_vector_type(8))) float v8f;

#define D 64
#define EPS_BN 1e-4f

// ---------------------------------------------------------------------------
// Utility: zero a float buffer
// ---------------------------------------------------------------------------
__global__ void zero_f32(float* __restrict__ p, int n) {
    int i = blockIdx.x * blockDim.x + threadIdx.x;
    if (i < n) p[i] = 0.0f;
}

// ---------------------------------------------------------------------------
// Degree: deg[dst] += 1 per edge
// ---------------------------------------------------------------------------
__global__ void deg_accum(const int* __restrict__ dst, float* __restrict__ deg, int E) {
    int e = blockIdx.x * blockDim.x + threadIdx.x;
    if (e < E) atomicAdd(&deg[dst[e]], 1.0f);
}

// dinv = rsqrt(deg+1), dinv2 = dinv*dinv  (matches reference exactly)
__global__ void deg_finalize(const float* __restrict__ deg, float* __restrict__ dinv,
                             float* __restrict__ dinv2, int N) {
    int i = blockIdx.x * blockDim.x + threadIdx.x;
    if (i < N) {
        float r = rsqrtf(deg[i] + 1.0f);
        dinv[i]  = r;
        dinv2[i] = r * r;
    }
}

// ---------------------------------------------------------------------------
// WMMA GEMM: H[N,64] = X[N,64] @ W[64,64]   (f32 WMMA, 16x16x4)
// Block = 128 threads = 4 waves; wave w handles column tile [16w,16w+16).
// Each block covers 128 rows (8 row-tiles of 16), B-fragments preloaded.
// ---------------------------------------------------------------------------
__global__ void gemm_wmma_f32(const float* __restrict__ X, const float* __restrict__ W,
                              float* __restrict__ H, int N) {
    const int wave = threadIdx.x >> 5;      // 0..3 -> column tile
    const int lane = threadIdx.x & 31;
    const int col0 = wave * 16;
    const int hi   = lane >> 4;             // 0: K pair {0,1}, 1: K pair {2,3}
    const int nn   = lane & 15;             // N (for B/C/D) or M (for A)
    const int rowBase = blockIdx.x * 128;

    // Preload B fragments for all 16 K-steps.
    // B 4x16 f32 layout (mirror of A): VGPR0 = row K=2*hi, VGPR1 = row K=2*hi+1, col = lane&15
    v2f bf[16];
#pragma unroll
    for (int s = 0; s < 16; ++s) {
        int k = 4 * s + 2 * hi;
        bf[s].x = W[(k + 0) * D + col0 + nn];
        bf[s].y = W[(k + 1) * D + col0 + nn];
    }

#pragma unroll 1
    for (int t = 0; t < 8; ++t) {
        int row0 = rowBase + t * 16;        // uniform across wave
        if (row0 >= N) break;               // uniform branch: EXEC stays all-ones
        int rA = row0 + nn;                 // A-matrix: M = lane&15
        if (rA >= N) rA = 0;                // clamp (store is guarded)

        v8f c = {};
#pragma unroll
        for (int s = 0; s < 16; ++s) {
            int k = 4 * s + 2 * hi;
            // A 16x4 f32: lanes 0-15 -> K={4s,4s+1}, lanes 16-31 -> K={4s+2,4s+3}
            v2f a = *reinterpret_cast<const v2f*>(&X[rA * D + k]);
            c = __builtin_amdgcn_wmma_f32_16x16x4_f32(
                    false, a, false, bf[s], (short)0, c, false, false);
        }
        // C/D 16x16 f32: VGPR r -> row (r + 8*hi), col = lane&15
#pragma unroll
        for (int r = 0; r < 8; ++r) {
            int row = row0 + r + 8 * hi;
            if (row < N) H[row * D + col0 + nn] = c[r];
        }
    }
}

// ---------------------------------------------------------------------------
// agg = h * dinv2[row] + b[col]   (self-loop term + bias folded in)
// ---------------------------------------------------------------------------
__global__ void init_agg(const float* __restrict__ h, const float* __restrict__ dinv2,
                         const float* __restrict__ b, float* __restrict__ agg, int total) {
    int i = blockIdx.x * blockDim.x + threadIdx.x;
    if (i < total) {
        int row = i >> 6, col = i & 63;
        agg[i] = h[i] * dinv2[row] + b[col];
    }
}

// ---------------------------------------------------------------------------
// Edge scatter: one wave (32 lanes) per edge; lane handles 2 features.
// agg[dst] += h[src] * dinv[src]*dinv[dst]
// ---------------------------------------------------------------------------
__global__ void edge_scatter(const float* __restrict__ h, const int* __restrict__ src,
                             const int* __restrict__ dst, const float* __restrict__ dinv,
                             float* __restrict__ agg, int E) {
    int gid  = blockIdx.x * blockDim.x + threadIdx.x;
    int e    = gid >> 5;
    int lane = gid & 31;
    if (e >= E) return;
    int s = src[e];
    int d = dst[e];
    float coeff = dinv[s] * dinv[d];
    const v2f hv = *reinterpret_cast<const v2f*>(&h[s * D + lane * 2]);
    atomicAdd(&agg[d * D + lane * 2 + 0], hv.x * coeff);
    atomicAdd(&agg[d * D + lane * 2 + 1], hv.y * coeff);
}

// ---------------------------------------------------------------------------
// BN pass 1: apply activation in-place, accumulate per-column sum / sumsq.
// mode 0: sigmoid(x)   mode 1: identity   mode 2: sigmoid(0.5x)
// 256 threads: col = tid&63, 4 row-lanes per column; grid-stride over rows.
// ---------------------------------------------------------------------------
__global__ void bn_reduce(float* __restrict__ a, float* __restrict__ stats,
                          int N, int mode) {
    __shared__ float ssum[256];
    __shared__ float ssq[256];
    int col = threadIdx.x & 63;
    int rq  = threadIdx.x >> 6;   // 0..3

    float s = 0.0f, q = 0.0f;
    for (int row = blockIdx.x * 4 + rq; row < N; row += gridDim.x * 4) {
        float v = a[row * D + col];
        if (mode == 0)      v = 1.0f / (1.0f + __expf(-v));
        else if (mode == 2) v = 1.0f / (1.0f + __expf(-0.5f * v));
        a[row * D + col] = v;
        s += v;
        q += v * v;
    }
    ssum[threadIdx.x] = s;
    ssq[threadIdx.x]  = q;
    __syncthreads();
    if (threadIdx.x < 64) {
        float ts = ssum[threadIdx.x] + ssum[threadIdx.x + 64] +
                   ssum[threadIdx.x + 128] + ssum[threadIdx.x + 192];
        float tq = ssq[threadIdx.x] + ssq[threadIdx.x + 64] +
                   ssq[threadIdx.x + 128] + ssq[threadIdx.x + 192];
        atomicAdd(&stats[threadIdx.x], ts);
        atomicAdd(&stats[64 + threadIdx.x], tq);
    }
}

// ---------------------------------------------------------------------------
// BN pass 2: out = (a - m) * rsqrt(var + eps) * gamma + beta  (biased var)
// ---------------------------------------------------------------------------
__global__ void bn_finalize(const float* __restrict__ a, const float* __restrict__ stats,
                            const float* __restrict__ gamma, const float* __restrict__ beta,
                            float* __restrict__ out, int N) {
    int i = blockIdx.x * blockDim.x + threadIdx.x;
    int total = N * D;
    if (i >= total) return;
    int col = i & 63;
    float invN = 1.0f / (float)N;
    float m = stats[col] * invN;
    float v = stats[64 + col] * invN - m * m;
    out[i] = (a[i] - m) * rsqrtf(v + EPS_BN) * gamma[col] + beta[col];
}

// ---------------------------------------------------------------------------
// Host-side: one full GCN layer (conv -> activation -> BN)
// ---------------------------------------------------------------------------
static void run_layer(const float* in, const float* W, const float* b,
                      const float* gamma, const float* beta, float* out, int mode,
                      const int* src, const int* dst,
                      float* h, float* agg, float* dinv, float* dinv2, float* stats,
                      int N, int E, hipStream_t stream) {
    int total = N * D;
    gemm_wmma_f32<<<(N + 127) / 128, 128, 0, stream>>>(in, W, h, N);
    init_agg<<<(total + 255) / 256, 256, 0, stream>>>(h, dinv2, b, agg, total);
    {
        long long work = (long long)E * 32;
        unsigned blocks = (unsigned)((work + 255) / 256);
        edge_scatter<<<blocks, 256, 0, stream>>>(h, src, dst, dinv, agg, E);
    }
    zero_f32<<<1, 128, 0, stream>>>(stats, 128);
    bn_reduce<<<1024, 256, 0, stream>>>(agg, stats, N, mode);
    bn_finalize<<<(total + 255) / 256, 256, 0, stream>>>(agg, stats, gamma, beta, out, N);
}

// ---------------------------------------------------------------------------
// Entry point
// ---------------------------------------------------------------------------
extern "C" void kernel_launch(void* const* d_in, const int* in_sizes, int n_in,
                              void* d_out, int out_size, void* d_ws, size_t ws_size,
                              hipStream_t stream) {
    const float* x   = (const float*)d_in[0];
    const int*   ei  = (const int*)d_in[1];
    // d_in[2] = batch (unused; pooled output discarded upstream)
    const float* enc_W  = (const float*)d_in[3];
    const float* enc_b  = (const float*)d_in[4];
    const float* enc_g  = (const float*)d_in[5];
    const float* enc_be = (const float*)d_in[6];
    const float* att_W  = (const float*)d_in[7];
    const float* att_b  = (const float*)d_in[8];
    const float* att_g  = (const float*)d_in[9];
    const float* att_be = (const float*)d_in[10];
    const float* str_W  = (const float*)d_in[11];
    const float* str_b  = (const float*)d_in[12];
    const float* str_g  = (const float*)d_in[13];
    const float* str_be = (const float*)d_in[14];

    const int N = in_sizes[0] / D;
    const int E = in_sizes[1] / 2;
    const int* src = ei;
    const int* dst = ei + E;

    // Workspace layout (floats)
    float* ws    = (float*)d_ws;
    size_t nd    = (size_t)N * D;
    float* h     = ws;
    float* agg   = h + nd;
    float* bufA  = agg + nd;
    float* deg   = bufA + nd;
    float* dinv  = deg + N;
    float* dinv2 = dinv + N;
    float* stats = dinv2 + N;   // 128 floats

    // Outputs: (x_str, x_att, x_enc) concatenated
    float* x_str = (float*)d_out;
    float* x_att = x_str + nd;
    float* x_enc = x_att + nd;

    // ---- degree / normalization factors (once) ----
    zero_f32<<<(N + 255) / 256, 256, 0, stream>>>(deg, N);
    deg_accum<<<(E + 255) / 256, 256, 0, stream>>>(dst, deg, E);
    deg_finalize<<<(N + 255) / 256, 256, 0, stream>>>(deg, dinv, dinv2, N);

    const int WOFF = D * D;   // per-layer weight offset
    // ---- Encoder: sigmoid + BN both layers ----
    run_layer(x,    enc_W,        enc_b,     enc_g,     enc_be,     bufA,  0,
              src, dst, h, agg, dinv, dinv2, stats, N, E, stream);
    run_layer(bufA, enc_W + WOFF, enc_b + D, enc_g + D, enc_be + D, x_enc, 0,
              src, dst, h, agg, dinv, dinv2, stats, N, E, stream);

    // ---- Attribute decoder: sigmoid on layer 0, identity on last ----
    run_layer(x_enc, att_W,        att_b,     att_g,     att_be,     bufA,  0,
              src, dst, h, agg, dinv, dinv2, stats, N, E, stream);
    run_layer(bufA,  att_W + WOFF, att_b + D, att_g + D, att_be + D, x_att, 1,
              src, dst, h, agg, dinv, dinv2, stats, N, E, stream);

    // ---- Structure decoder: sigmoid on layer 0, sigmoid(0.5x) on last ----
    run_layer(x_enc, str_W,        str_b,     str_g,     str_be,     bufA,  0,
              src, dst, h, agg, dinv, dinv2, stats, N, E, stream);
    run_layer(bufA,  str_W + WOFF, str_b + D, str_g + D, str_be + D, x_str, 2,
              src, dst, h, agg, dinv, dinv2, stats, N, E, stream);
}